// BatchTopK_42271068127405
// MI455X (gfx1250) — compile-verified
//
#include <hip/hip_runtime.h>
#include <hip/hip_bf16.h>
#include <stdint.h>

// ---------------------------------------------------------------------------
// Global Batch-TopK (keep top k*rows relu'd values of the whole tensor).
// Exact 2-level radix select on float bit patterns (positive floats order as
// unsigned ints). CDNA5 paths used: async global->LDS DMA (ASYNCcnt),
// LDS atomics, wave32 sizing.
// ---------------------------------------------------------------------------

#define NB1 32768   // high-16-bit buckets (positive finite floats => idx < 32640)
#define NB2 65536   // low-16-bit buckets
#define TPB_HIST 1024

__device__ __forceinline__ uint32_t lds_offset(const void* p) {
    // Generic (flat) pointers to LDS encode the wave-relative LDS byte offset
    // in their low 32 bits (ISA 10.2 aperture mapping).
    return (uint32_t)(uintptr_t)p;
}

__device__ __forceinline__ void async_load_b128(uint32_t lds_addr, const void* gaddr) {
    asm volatile("global_load_async_to_lds_b128 %0, %1, off"
                 :: "v"(lds_addr), "v"((unsigned long long)(uintptr_t)gaddr)
                 : "memory");
}

// -------------------------- zero scratch -----------------------------------
__global__ void zero_kernel(unsigned* __restrict__ ws, int n) {
    int i = blockIdx.x * blockDim.x + threadIdx.x;
    if (i < n) ws[i] = 0u;
}

// ------------------- Pass A: high-16-bit histogram (async staged) ----------
__global__ __launch_bounds__(TPB_HIST)
void hist_hi_kernel(const float* __restrict__ x, unsigned* __restrict__ hist1,
                    long long n4) {
    __shared__ unsigned lh[NB1];           // 128 KB
    __shared__ float4 stage[2][TPB_HIST];  //  32 KB

    for (int i = threadIdx.x; i < NB1; i += TPB_HIST) lh[i] = 0u;
    __syncthreads();

    const float4* x4 = (const float4*)x;
    const long long stride = (long long)gridDim.x * TPB_HIST;
    long long i = (long long)blockIdx.x * TPB_HIST + threadIdx.x;

    const uint32_t lds0 = lds_offset(&stage[0][threadIdx.x]);
    const uint32_t lds1 = lds_offset(&stage[1][threadIdx.x]);
    int p = 0;

    if (i < n4) async_load_b128(lds0, x4 + i);

    for (; i < n4; i += stride) {
        long long nxt = i + stride;
        if (nxt < n4) {
            async_load_b128(p ? lds0 : lds1, x4 + nxt);
            asm volatile("s_wait_asynccnt 1" ::: "memory");  // current buffer ready
        } else {
            asm volatile("s_wait_asynccnt 0" ::: "memory");
        }
        float4 v = stage[p][threadIdx.x];   // ds_load_b128
        float f0 = v.x, f1 = v.y, f2 = v.z, f3 = v.w;
        if (f0 > 0.0f) atomicAdd(&lh[__float_as_uint(f0) >> 16], 1u);
        if (f1 > 0.0f) atomicAdd(&lh[__float_as_uint(f1) >> 16], 1u);
        if (f2 > 0.0f) atomicAdd(&lh[__float_as_uint(f2) >> 16], 1u);
        if (f3 > 0.0f) atomicAdd(&lh[__float_as_uint(f3) >> 16], 1u);
        p ^= 1;
    }
    __syncthreads();
    for (int b = threadIdx.x; b < NB1; b += TPB_HIST) {
        unsigned c = lh[b];
        if (c) atomicAdd(&hist1[b], c);
    }
}

// ------------------- Scan high histogram (single block) --------------------
// scal layout: [0]=B_hi (0xFFFFFFFF sentinel), [1]=count_gt_hi,
//              [2]=threshold_bits, [3]=tie_budget, [4]=tie_counter, [5]=nkeep
__global__ __launch_bounds__(1024)
void scan_hi_kernel(const unsigned* __restrict__ hist1, const int* __restrict__ kptr,
                    long long rows, unsigned* __restrict__ scal) {
    __shared__ unsigned csum[1024];
    const int t = threadIdx.x;
    unsigned s = 0;
    #pragma unroll
    for (int j = 0; j < NB1 / 1024; ++j) s += hist1[t * (NB1 / 1024) + j];
    csum[t] = s;
    __syncthreads();
    if (t == 0) {
        long long nkeep = (long long)kptr[0] * rows;
        long long acc = 0;
        int bh = -1;
        long long cg = 0;
        for (int c = 1023; c >= 0; --c) {
            if (acc + (long long)csum[c] >= nkeep) {
                for (int b = c * (NB1 / 1024) + (NB1 / 1024) - 1; b >= c * (NB1 / 1024); --b) {
                    long long h = hist1[b];
                    if (acc + h >= nkeep) { bh = b; cg = acc; break; }
                    acc += h;
                }
                break;
            }
            acc += csum[c];
        }
        if (bh < 0 || nkeep <= 0) {
            // total positives <= nkeep (keep everything positive), or nkeep==0
            scal[0] = 0xFFFFFFFFu;
            scal[1] = 0u;
        } else {
            scal[0] = (unsigned)bh;
            scal[1] = (unsigned)cg;
        }
        scal[5] = (unsigned)(nkeep < 0 ? 0 : nkeep);
    }
}

// ------------------- Pass B: low-16-bit refine histogram -------------------
__global__ void hist_lo_kernel(const float* __restrict__ x,
                               const unsigned* __restrict__ scal,
                               unsigned* __restrict__ hist2, long long n4) {
    const unsigned bhi = scal[0];  // uniform scalar load; sentinel never matches
    const float4* x4 = (const float4*)x;
    const long long stride = (long long)gridDim.x * blockDim.x;
    for (long long i = (long long)blockIdx.x * blockDim.x + threadIdx.x; i < n4; i += stride) {
        float4 v = x4[i];
        float f[4] = {v.x, v.y, v.z, v.w};
        #pragma unroll
        for (int c = 0; c < 4; ++c) {
            if (f[c] > 0.0f) {
                unsigned bits = __float_as_uint(f[c]);
                if ((bits >> 16) == bhi) atomicAdd(&hist2[bits & 0xFFFFu], 1u);
            }
        }
    }
}

// ------------------- Scan low histogram (single block) ---------------------
__global__ __launch_bounds__(1024)
void scan_lo_kernel(const unsigned* __restrict__ hist2, unsigned* __restrict__ scal) {
    __shared__ unsigned csum[1024];
    const int t = threadIdx.x;
    unsigned s = 0;
    #pragma unroll
    for (int j = 0; j < NB2 / 1024; ++j) s += hist2[t * (NB2 / 1024) + j];
    csum[t] = s;
    __syncthreads();
    if (t == 0) {
        scal[4] = 0u;  // tie counter
        if (scal[0] == 0xFFFFFFFFu) {
            unsigned nkeep = scal[5];
            if (nkeep == 0u) { scal[2] = 0x7FFFFFFFu; scal[3] = 0u; }  // keep nothing
            else             { scal[2] = 0u;          scal[3] = 0u; }  // keep all positives
            return;
        }
        unsigned nkeep = scal[5];
        long long acc = scal[1];
        for (int c = 1023; c >= 0; --c) {
            if (acc + (long long)csum[c] >= (long long)nkeep) {
                for (int l = c * (NB2 / 1024) + (NB2 / 1024) - 1; l >= c * (NB2 / 1024); --l) {
                    long long h = hist2[l];
                    if (acc + h >= (long long)nkeep) {
                        scal[2] = (scal[0] << 16) | (unsigned)l;
                        scal[3] = (unsigned)((long long)nkeep - acc);
                        return;
                    }
                    acc += h;
                }
            }
            acc += csum[c];
        }
    }
}

// ------------------- Pass C: threshold + scatter-back ----------------------
__global__ void select_kernel(const float* __restrict__ x, float* __restrict__ out,
                              unsigned* scal, long long n4) {
    const unsigned thr = scal[2];
    const unsigned budget = scal[3];
    const float4* x4 = (const float4*)x;
    float4* o4 = (float4*)out;
    const long long stride = (long long)gridDim.x * blockDim.x;
    for (long long i = (long long)blockIdx.x * blockDim.x + threadIdx.x; i < n4; i += stride) {
        float4 v = x4[i];
        float f[4] = {v.x, v.y, v.z, v.w};
        float o[4];
        #pragma unroll
        for (int c = 0; c < 4; ++c) {
            float r = 0.0f;
            if (f[c] > 0.0f) {
                unsigned bits = __float_as_uint(f[c]);
                if (bits > thr) {
                    r = f[c];
                } else if (bits == thr) {
                    unsigned pos = atomicAdd(&scal[4], 1u);  // ties: measure-zero for real data
                    if (pos < budget) r = f[c];
                }
            }
            o[c] = r;
        }
        float4 w; w.x = o[0]; w.y = o[1]; w.z = o[2]; w.w = o[3];
        o4[i] = w;
    }
}

// ---------------------------------------------------------------------------
extern "C" void kernel_launch(void* const* d_in, const int* in_sizes, int n_in,
                              void* d_out, int out_size, void* d_ws, size_t ws_size,
                              hipStream_t stream) {
    const float* x = (const float*)d_in[0];
    const int* kptr = (const int*)d_in[1];

    const long long N = (long long)in_sizes[0];
    const long long rows = N / 16384;  // product of all-but-last dims (2048)
    const long long n4 = N / 4;

    unsigned* hist1 = (unsigned*)d_ws;        // 32768
    unsigned* hist2 = hist1 + NB1;            // 65536
    unsigned* scal  = hist2 + NB2;            // 8

    const int zn = NB1 + NB2 + 8;
    zero_kernel<<<(zn + 255) / 256, 256, 0, stream>>>((unsigned*)d_ws, zn);

    hist_hi_kernel<<<512, TPB_HIST, 0, stream>>>(x, hist1, n4);
    scan_hi_kernel<<<1, 1024, 0, stream>>>(hist1, kptr, rows, scal);
    hist_lo_kernel<<<4096, 256, 0, stream>>>(x, scal, hist2, n4);
    scan_lo_kernel<<<1, 1024, 0, stream>>>(hist2, scal);
    select_kernel<<<4096, 256, 0, stream>>>(x, (float*)d_out, scal, n4);
}